// MemoryAccess_21852793602124
// MI455X (gfx1250) — compile-verified
//
#include <hip/hip_runtime.h>
#include <math.h>

// DNC memory access, MI455X (gfx1250, wave32).
// Sizes from reference: B=1024, D=256, M=256, W=64, R=4, NW=1, TOT=471.
#define B_   1024
#define D_   256
#define M_   256
#define W_   64
#define R_   4
#define TOT_ 471
#define OST_ 480      // padded row stride for the projection output `o`
#define EPS_ 1e-6f
#define LSTR 260      // link-stripe LDS stride: %4==0 (b128) and bank-spread
#define PSTR 257      // read-weights LDS stride (conflict-free A-tile reads)

typedef float v2f __attribute__((ext_vector_type(2)));
typedef float v8f __attribute__((ext_vector_type(8)));

__device__ __forceinline__ float sigm(float x)  { return 1.0f / (1.0f + expf(-x)); }
__device__ __forceinline__ float softpl(float x){ return x > 20.0f ? x : log1pf(expf(x)); }

// ---------------------------------------------------------------------------
// K1: o = inputs @ W_all + b_all   (1024 x 471, K=256) via V_WMMA_F32_16X16X4_F32
// One wave per 16x16 output tile. A: 16x4 f32 (lane=M row, VGPR0/1 = K pair,
// lanes 16-31 hold K+2). B: 4x16 (VGPR v: K=v for lanes 0-15, K=v+2 for 16-31).
// C/D: VGPR v -> row v (+8 for upper half-lanes), col = lane&15.
// ---------------------------------------------------------------------------
__global__ void k1_gemm(const float* __restrict__ A, const float* __restrict__ Wm,
                        const float* __restrict__ bias, float* __restrict__ o)
{
    const int lane = threadIdx.x & 31;
    const int half = lane >> 4, l16 = lane & 15;
    const int rb = blockIdx.x, cb = blockIdx.y;
    const int row  = rb * 16 + l16;
    const int col  = cb * 16 + l16;
    const int colc = (col < TOT_) ? col : (TOT_ - 1);

    v8f c = {};
    for (int k0 = 0; k0 < D_; k0 += 4) {
        v2f a, bm;
        a.x  = A[row * D_ + k0 + 2 * half];
        a.y  = A[row * D_ + k0 + 2 * half + 1];
        bm.x = Wm[(k0 + 2 * half) * TOT_ + colc];
        bm.y = Wm[(k0 + 2 * half + 1) * TOT_ + colc];
        c = __builtin_amdgcn_wmma_f32_16x16x4_f32(false, a, false, bm,
                                                  (short)0, c, false, false);
    }
    if (col < TOT_) {
        float bv = bias[col];
        for (int v = 0; v < 8; ++v) {
            int orow = rb * 16 + v + 8 * half;
            o[orow * OST_ + col] = c[v] + bv;
        }
    }
}

// ---------------------------------------------------------------------------
// K2: usage = (pu + (1-pu)*pw) * prod_r(1 - sigmoid(free_r)*prev_read[r,m])
// ---------------------------------------------------------------------------
__global__ void k2_usage(const float* __restrict__ o, const float* __restrict__ prev_ww,
                         const float* __restrict__ prev_rw, const float* __restrict__ prev_usage,
                         float* __restrict__ usage_out)
{
    int b = blockIdx.x, m = threadIdx.x;
    float pu = prev_usage[b * M_ + m];
    float pw = prev_ww[b * M_ + m];            // NW==1: ww_agg == prev_write_weights
    float u  = pu + (1.0f - pu) * pw;
    float phi = 1.0f;
    for (int r = 0; r < R_; ++r) {
        float fg = sigm(o[b * OST_ + 128 + r]);
        phi *= (1.0f - fg * prev_rw[b * R_ * M_ + r * M_ + m]);
    }
    usage_out[b * M_ + m] = u * phi;
}

// ---------------------------------------------------------------------------
// K3: write content weights = softmax_m( strength * cos(write_key, prev_mem[m]) )
// ---------------------------------------------------------------------------
__global__ void k3_wcw(const float* __restrict__ o, const float* __restrict__ prev_mem,
                       float* __restrict__ wcw)
{
    __shared__ float wk[W_];
    __shared__ float red[M_];
    int b = blockIdx.x, t = threadIdx.x;
    if (t < W_) wk[t] = o[b * OST_ + 151 + t];
    __syncthreads();

    float kn = 0.f, dot = 0.f, mn = 0.f;
    for (int w = 0; w < W_; ++w) {
        float kv = wk[w], mv = prev_mem[(long)b * M_ * W_ + t * W_ + w];
        kn += kv * kv; dot += kv * mv; mn += mv * mv;
    }
    kn = sqrtf(kn); mn = sqrtf(mn);
    float s   = softpl(o[b * OST_ + 146]);
    float sim = dot / (kn * mn + EPS_) * s;

    red[t] = sim; __syncthreads();
    for (int off = 128; off > 0; off >>= 1) { if (t < off) red[t] = fmaxf(red[t], red[t + off]); __syncthreads(); }
    float mx = red[0]; __syncthreads();
    float e = expf(sim - mx); red[t] = e; __syncthreads();
    for (int off = 128; off > 0; off >>= 1) { if (t < off) red[t] += red[t + off]; __syncthreads(); }
    wcw[b * M_ + t] = e / red[0];
}

// ---------------------------------------------------------------------------
// K4: allocation weighting (bitonic sort + multiplicative exclusive scan +
// inverse-permutation scatter), then write_weights and precedence.
// ---------------------------------------------------------------------------
__global__ void k4_alloc(const float* __restrict__ o, const float* __restrict__ wcw,
                         const float* __restrict__ usage, const float* __restrict__ prev_prec,
                         float* __restrict__ ww_out, float* __restrict__ prec_out)
{
    __shared__ float key_s[M_];
    __shared__ int   idx_s[M_];
    __shared__ float sc_s[M_];
    __shared__ float al_s[M_];
    __shared__ float red[M_];
    int b = blockIdx.x, t = threadIdx.x;

    float u = usage[b * M_ + t];
    u = EPS_ + (1.0f - EPS_) * u;
    key_s[t] = 1.0f - u;     // nonusage, sort descending
    idx_s[t] = t;
    __syncthreads();

    for (int k = 2; k <= M_; k <<= 1) {
        for (int j = k >> 1; j > 0; j >>= 1) {
            int ixj = t ^ j;
            if (ixj > t) {
                float a = key_s[t], c = key_s[ixj];
                bool desc = ((t & k) == 0);
                if (desc ? (a < c) : (a > c)) {
                    key_s[t] = c; key_s[ixj] = a;
                    int ia = idx_s[t]; idx_s[t] = idx_s[ixj]; idx_s[ixj] = ia;
                }
            }
            __syncthreads();
        }
    }
    float ex = (t == 0) ? 1.0f : (1.0f - key_s[t - 1]);
    sc_s[t] = ex;
    __syncthreads();
    for (int off = 1; off < M_; off <<= 1) {
        float v = (t >= off) ? sc_s[t - off] : 1.0f;
        __syncthreads();
        sc_s[t] *= v;
        __syncthreads();
    }
    al_s[idx_s[t]] = key_s[t] * sc_s[t];
    __syncthreads();

    float ag = sigm(o[b * OST_ + 132]);
    float wg = sigm(o[b * OST_ + 133]);
    float ww = wg * (ag * al_s[t] + (1.0f - ag) * wcw[b * M_ + t]);
    ww_out[b * M_ + t] = ww;

    red[t] = ww; __syncthreads();
    for (int off = 128; off > 0; off >>= 1) { if (t < off) red[t] += red[t + off]; __syncthreads(); }
    float S = red[0];
    prec_out[b * M_ + t] = (1.0f - S) * prev_prec[b * M_ + t] + ww;
}

// ---------------------------------------------------------------------------
// K5: memory = prev_mem * (1 - ww[m]*erase[w]) + ww[m]*write_vec[w]   (NW==1)
// float4-vectorized (16.8M elements, pure HBM stream).
// ---------------------------------------------------------------------------
__global__ void k5_mem(const float* __restrict__ o, const float* __restrict__ ww,
                       const float* __restrict__ prev_mem, float* __restrict__ mem_out)
{
    long i4 = ((long)blockIdx.x * blockDim.x + threadIdx.x) * 4;
    if (i4 >= (long)B_ * M_ * W_) return;
    int w = (int)(i4 & (W_ - 1));
    int m = (int)((i4 >> 6) & (M_ - 1));
    int b = (int)(i4 >> 14);
    float4 e4  = *(const float4*)&o[b * OST_ + 64 + w];
    float4 wv4 = *(const float4*)&o[b * OST_ + w];
    float  wwv = ww[b * M_ + m];
    float4 pm  = *(const float4*)&prev_mem[i4];
    float4 r;
    r.x = pm.x * (1.0f - wwv * sigm(e4.x)) + wwv * wv4.x;
    r.y = pm.y * (1.0f - wwv * sigm(e4.y)) + wwv * wv4.y;
    r.z = pm.z * (1.0f - wwv * sigm(e4.z)) + wwv * wv4.z;
    r.w = pm.w * (1.0f - wwv * sigm(e4.w)) + wwv * wv4.w;
    *(float4*)&mem_out[i4] = r;
}

// ---------------------------------------------------------------------------
// K6 (fused, bandwidth-critical): link update + fwd/bwd read weights.
//   link[m,n] = (1 - w[m] - w[n])*prev_link[m,n] + w[m]*p[n], diag=0
//   bwd[r,n]  = sum_m pr[r,m]*link[m,n]   (persistent WMMA accumulators)
//   fwd[r,m]  = sum_n pr[r,n]*link[m,n]   (per-stripe WMMA + LDS reduce)
// prev_link streamed ONCE into double-buffered LDS stripes using the CDNA5
// async path (GLOBAL_LOAD_ASYNC_TO_LDS_B128 + s_wait_asynccnt): the next
// stripe's HBM fetch overlaps the current stripe's transform + WMMA.
// link written once with b128 stores. ~540 MB total HBM traffic @ 23.3 TB/s.
// ---------------------------------------------------------------------------
__global__ void k6_link(const float* __restrict__ ww, const float* __restrict__ prev_prec,
                        const float* __restrict__ prev_link, const float* __restrict__ prev_rw,
                        float* __restrict__ link_out, float* __restrict__ fwd,
                        float* __restrict__ bwd)
{
    __shared__ float pr_s[16 * PSTR];   // read weights padded to 16 rows
    __shared__ float w_s[M_];
    __shared__ float p_s[M_];
    __shared__ float L[2][16 * LSTR];   // double-buffered link stripe (padded rows)
    __shared__ float fpart[8 * 256];    // per-wave fwd partial tiles

    const int b = blockIdx.x, t = threadIdx.x;
    const int wave = t >> 5, lane = t & 31, half = lane >> 4, l16 = lane & 15;

    w_s[t] = ww[b * M_ + t];
    p_s[t] = prev_prec[b * M_ + t];
    for (int r = 0; r < 16; ++r)
        pr_s[r * PSTR + t] = (r < R_) ? prev_rw[b * R_ * M_ + r * M_ + t] : 0.0f;

    // async copy setup: 16 threads per stripe row, 16B per lane, 4 passes
    const unsigned long long gbase =
        (unsigned long long)(size_t)(const void*)(prev_link + (long)b * M_ * M_);
    const unsigned lds0 = (unsigned)(size_t)(const void*)&L[0][0];
    const unsigned lds1 = (unsigned)(size_t)(const void*)&L[1][0];
    const int arow = t >> 4;           // 0..15 stripe row
    const int acol = (t & 15) * 16;    // byte chunk within 256B sub-block

    // prologue: bring stripe 0 into L[0]
    for (int p = 0; p < 4; ++p) {
        unsigned la = lds0 + (unsigned)(arow * (LSTR * 4) + acol + p * 256);
        unsigned vo = (unsigned)(arow * 1024 + acol + p * 256);
        asm volatile("global_load_async_to_lds_b128 %0, %1, %2"
                     :: "v"(la), "v"(vo), "s"(gbase) : "memory");
    }
    __syncthreads();   // also covers pr_s / w_s / p_s

    v8f cb0 = {}, cb1 = {};
    const int n0a = (2 * wave) * 16, n0b = (2 * wave + 1) * 16;
    const int rowgrp = t >> 6;          // 0..3 (4 rows each in transform)
    const int col4   = (t & 63) * 4;    // 4-column group

    for (int m0 = 0; m0 < M_; m0 += 16) {
        const int buf = (m0 >> 4) & 1;
        float* Ls = &L[buf][0];

        if (m0 + 16 < M_) {            // issue next stripe into the other buffer
            const unsigned ldsn = buf ? lds0 : lds1;
            for (int p = 0; p < 4; ++p) {
                unsigned la = ldsn + (unsigned)(arow * (LSTR * 4) + acol + p * 256);
                unsigned vo = (unsigned)((m0 + 16) * 1024 + arow * 1024 + acol + p * 256);
                asm volatile("global_load_async_to_lds_b128 %0, %1, %2"
                             :: "v"(la), "v"(vo), "s"(gbase) : "memory");
            }
            asm volatile("s_wait_asynccnt 0x4" ::: "memory");  // current stripe done
        } else {
            asm volatile("s_wait_asynccnt 0x0" ::: "memory");
        }
        __syncthreads();

        // in-place transform prev_link stripe -> link stripe; b128 store to HBM
        for (int jj = 0; jj < 4; ++jj) {
            int j = rowgrp * 4 + jj;
            int m = m0 + j;
            float4 pl = *(float4*)&Ls[j * LSTR + col4];
            float wm = w_s[m];
            float4 v;
            v.x = (1.0f - wm - w_s[col4 + 0]) * pl.x + wm * p_s[col4 + 0];
            v.y = (1.0f - wm - w_s[col4 + 1]) * pl.y + wm * p_s[col4 + 1];
            v.z = (1.0f - wm - w_s[col4 + 2]) * pl.z + wm * p_s[col4 + 2];
            v.w = (1.0f - wm - w_s[col4 + 3]) * pl.w + wm * p_s[col4 + 3];
            if (m == col4 + 0) v.x = 0.0f;
            if (m == col4 + 1) v.y = 0.0f;
            if (m == col4 + 2) v.z = 0.0f;
            if (m == col4 + 3) v.w = 0.0f;
            *(float4*)&Ls[j * LSTR + col4] = v;
            *(float4*)&link_out[(long)b * M_ * M_ + m * M_ + col4] = v;
        }
        __syncthreads();

        // bwd: each wave owns two 16-wide n-tiles, accumulates over all stripes
        for (int kk = 0; kk < 16; kk += 4) {
            v2f a;
            a.x = pr_s[l16 * PSTR + m0 + kk + 2 * half];
            a.y = pr_s[l16 * PSTR + m0 + kk + 2 * half + 1];
            v2f b0, b1;
            b0.x = Ls[(kk + 2 * half) * LSTR + n0a + l16];
            b0.y = Ls[(kk + 2 * half + 1) * LSTR + n0a + l16];
            cb0 = __builtin_amdgcn_wmma_f32_16x16x4_f32(false, a, false, b0,
                                                        (short)0, cb0, false, false);
            b1.x = Ls[(kk + 2 * half) * LSTR + n0b + l16];
            b1.y = Ls[(kk + 2 * half + 1) * LSTR + n0b + l16];
            cb1 = __builtin_amdgcn_wmma_f32_16x16x4_f32(false, a, false, b1,
                                                        (short)0, cb1, false, false);
        }

        // fwd: wave handles K-chunk n in [32*wave, 32*wave+32); B = stripe^T
        v8f cf = {};
        for (int kk = 0; kk < 32; kk += 4) {
            int n = 32 * wave + kk;
            v2f a, bb;
            a.x  = pr_s[l16 * PSTR + n + 2 * half];
            a.y  = pr_s[l16 * PSTR + n + 2 * half + 1];
            bb.x = Ls[l16 * LSTR + n + 2 * half];     // transposed, conflict-free
            bb.y = Ls[l16 * LSTR + n + 2 * half + 1];
            cf = __builtin_amdgcn_wmma_f32_16x16x4_f32(false, a, false, bb,
                                                       (short)0, cf, false, false);
        }
        for (int v = 0; v < 8; ++v)
            fpart[wave * 256 + (v + 8 * half) * 16 + l16] = cf[v];
        __syncthreads();

        float s = 0.0f;
        for (int wv = 0; wv < 8; ++wv) s += fpart[wv * 256 + t];
        if ((t >> 4) < R_)
            fwd[b * R_ * M_ + (t >> 4) * M_ + m0 + (t & 15)] = s;
        __syncthreads();
    }

    if (half == 0) {
        for (int v = 0; v < R_; ++v) {
            bwd[b * R_ * M_ + v * M_ + n0a + l16] = cb0[v];
            bwd[b * R_ * M_ + v * M_ + n0b + l16] = cb1[v];
        }
    }
}

// ---------------------------------------------------------------------------
// K7: read content weights (cosine vs NEW memory, softmax), combine with
// fwd/bwd via read-mode softmax, then read_words = read_weights @ memory.
// ---------------------------------------------------------------------------
__global__ void k7_read(const float* __restrict__ o, const float* __restrict__ mem,
                        const float* __restrict__ fwd, const float* __restrict__ bwd,
                        float* __restrict__ rw_out, float* __restrict__ words_out)
{
    __shared__ float rk_s[R_ * W_];
    __shared__ float rw_s[R_ * M_];
    __shared__ float red[M_];
    int b = blockIdx.x, t = threadIdx.x;

    rk_s[t] = o[b * OST_ + 215 + t];
    __syncthreads();

    int m = t;
    float nm = 0.f;
    for (int w = 0; w < W_; ++w) { float v = mem[(long)b * M_ * W_ + m * W_ + w]; nm += v * v; }
    nm = sqrtf(nm);

    float sim[R_];
    for (int r = 0; r < R_; ++r) {
        float kn = 0.f, dot = 0.f;
        for (int w = 0; w < W_; ++w) {
            float kv = rk_s[r * W_ + w];
            kn  += kv * kv;
            dot += kv * mem[(long)b * M_ * W_ + m * W_ + w];
        }
        kn = sqrtf(kn);
        sim[r] = dot / (kn * nm + EPS_) * softpl(o[b * OST_ + 147 + r]);
    }

    float cw[R_];
    for (int r = 0; r < R_; ++r) {
        red[t] = sim[r]; __syncthreads();
        for (int off = 128; off > 0; off >>= 1) { if (t < off) red[t] = fmaxf(red[t], red[t + off]); __syncthreads(); }
        float mx = red[0]; __syncthreads();
        float e = expf(sim[r] - mx); red[t] = e; __syncthreads();
        for (int off = 128; off > 0; off >>= 1) { if (t < off) red[t] += red[t + off]; __syncthreads(); }
        cw[r] = e / red[0];
        __syncthreads();
    }

    for (int r = 0; r < R_; ++r) {
        float e0 = o[b * OST_ + 134 + 3 * r];
        float e1 = o[b * OST_ + 134 + 3 * r + 1];
        float e2 = o[b * OST_ + 134 + 3 * r + 2];
        float mx = fmaxf(e0, fmaxf(e1, e2));
        float x0 = expf(e0 - mx), x1 = expf(e1 - mx), x2 = expf(e2 - mx);
        float inv = 1.0f / (x0 + x1 + x2);
        float bm = x0 * inv, fm = x1 * inv, cm = x2 * inv;
        float rv = cm * cw[r]
                 + fm * fwd[b * R_ * M_ + r * M_ + m]
                 + bm * bwd[b * R_ * M_ + r * M_ + m];
        rw_s[r * M_ + m] = rv;
        rw_out[b * R_ * M_ + r * M_ + m] = rv;
    }
    __syncthreads();

    int rr = t >> 6, wcol = t & 63;
    float acc = 0.f;
    for (int mm = 0; mm < M_; ++mm)
        acc += rw_s[rr * M_ + mm] * mem[(long)b * M_ * W_ + mm * W_ + wcol];
    words_out[b * R_ * W_ + t] = acc;
}

// ---------------------------------------------------------------------------
extern "C" void kernel_launch(void* const* d_in, const int* in_sizes, int n_in,
                              void* d_out, int out_size, void* d_ws, size_t ws_size,
                              hipStream_t stream)
{
    (void)in_sizes; (void)n_in; (void)out_size; (void)ws_size;

    const float* inputs    = (const float*)d_in[0];
    const float* W_all     = (const float*)d_in[1];
    const float* b_all     = (const float*)d_in[2];
    const float* prev_mem  = (const float*)d_in[3];
    const float* prev_rw   = (const float*)d_in[4];
    const float* prev_ww   = (const float*)d_in[5];
    const float* prev_link = (const float*)d_in[6];
    const float* prev_prec = (const float*)d_in[7];
    const float* prev_use  = (const float*)d_in[8];

    float* out     = (float*)d_out;
    float* O_words = out;                               // B*R*W   = 262144
    float* O_mem   = O_words + 262144;                  // B*M*W   = 16777216
    float* O_rw    = O_mem   + 16777216;                // B*R*M   = 1048576
    float* O_ww    = O_rw    + 1048576;                 // B*M     = 262144
    float* O_link  = O_ww    + 262144;                  // B*M*M   = 67108864
    float* O_prec  = O_link  + 67108864;                // B*M     = 262144
    float* O_use   = O_prec  + 262144;                  // B*M     = 262144

    float* ws      = (float*)d_ws;
    float* ws_o    = ws;                                // B*OST   = 491520
    float* ws_wcw  = ws_o   + (size_t)B_ * OST_;        // B*M     = 262144
    float* ws_fwd  = ws_wcw + (size_t)B_ * M_;          // B*R*M   = 1048576
    float* ws_bwd  = ws_fwd + (size_t)B_ * R_ * M_;     // B*R*M   = 1048576

    k1_gemm <<<dim3(B_ / 16, (TOT_ + 15) / 16), 32, 0, stream>>>(inputs, W_all, b_all, ws_o);
    k2_usage<<<B_, M_, 0, stream>>>(ws_o, prev_ww, prev_rw, prev_use, O_use);
    k3_wcw  <<<B_, M_, 0, stream>>>(ws_o, prev_mem, ws_wcw);
    k4_alloc<<<B_, M_, 0, stream>>>(ws_o, ws_wcw, O_use, prev_prec, O_ww, O_prec);
    k5_mem  <<<(B_ * M_ * W_ / 4) / 256, 256, 0, stream>>>(ws_o, O_ww, prev_mem, O_mem);
    k6_link <<<B_, M_, 0, stream>>>(O_ww, prev_prec, prev_link, prev_rw, O_link, ws_fwd, ws_bwd);
    k7_read <<<B_, M_, 0, stream>>>(ws_o, O_mem, ws_fwd, ws_bwd, O_rw, O_words);
}